// FourierBlock_21612275434078
// MI455X (gfx1250) — compile-verified
//
#include <hip/hip_runtime.h>

// FourierBlock for MI455X (gfx1250): rfft->mode-mix->irfft rewritten as three
// batched fp32 GEMMs on v_wmma_f32_16x16x4_f32.
// B=64, L=1024, H=8, E=64, MODES=64. keys/values/mask are unused by the reference.
//
// Round-2 change: all operand tensors under our control (basis tables, repacked
// W, sel/out_ft intermediates) are stored in WMMA *fragment order*:
//   pos = ((K>>2)*2 + ((K>>1)&1)) * ncols + col , component = K&1
// so each lane's (j=0,j=1) fragment pair is one aligned, fully coalesced
// global_load_b64 instead of two b32 loads. Workspace use: ~52 MB.

#define B_ 64
#define L_ 1024
#define H_ 8
#define E_ 64
#define M_ 64   // MODES

typedef __attribute__((ext_vector_type(2))) float v2f;
typedef __attribute__((ext_vector_type(8))) float v8f;

// D = A(16x4) * B(4x16) + C, fp32. 8-arg form: (neg_a, A, neg_b, B, c_mod, C, reuse_a, reuse_b)
__device__ __forceinline__ v8f wmma_f32(v2f a, v2f b, v8f c) {
  return __builtin_amdgcn_wmma_f32_16x16x4_f32(false, a, false, b, (short)0, c, false, false);
}

// fragment-order index helper: K-index k, column col, row pitch ncols (in v2f units)
__device__ __forceinline__ int frag_pos(int k, int col, int ncols) {
  return ((k >> 2) * 2 + ((k >> 1) & 1)) * ncols + col;   // v2f element index
}

// ---------------------------------------------------------------------------
// Kernel 0: DFT basis tables, stored in fragment order.
// Forward  (A of stage 1): K = l (0..1023), col = w (64 cols).
//   Fc =  cos(2*pi*w*l/L), Fs = -sin(...)
// Inverse  (A of stage 3): K = w (0..63),  col = l (1024 cols).
//   Bc = s_w*cos, Bs = -s_w*sin, s_w = (w?2:1)/L
// Phase reduced exactly: (w*l) mod 1024.
// ---------------------------------------------------------------------------
__global__ void __launch_bounds__(256)
build_basis(float* __restrict__ Fc, float* __restrict__ Fs,
            float* __restrict__ Bc, float* __restrict__ Bs)
{
  const int idx = blockIdx.x * blockDim.x + threadIdx.x;  // 0..65535
  const int l = idx >> 6;
  const int w = idx & 63;
  const int ph = (w * l) & (L_ - 1);
  const float theta = (float)ph * (6.28318530717958647692f / (float)L_);
  float s, c;
  sincosf(theta, &s, &c);

  const int pf = frag_pos(l, w, M_) * 2 + (l & 1);        // forward: K=l
  Fc[pf] = c;
  Fs[pf] = -s;

  const float sc = (w == 0 ? 1.0f : 2.0f) / (float)L_;
  const int pi = frag_pos(w, l, L_) * 2 + (w & 1);        // inverse: K=w
  Bc[pi] = sc * c;
  Bs[pi] = -sc * s;
}

// ---------------------------------------------------------------------------
// Kernel 0b: repack W from (h,i,o,w) into per-(h,w) slabs in fragment order
// (B operand of stage 2: K = i, col = o).
// ---------------------------------------------------------------------------
__global__ void __launch_bounds__(256)
transpose_w(const float* __restrict__ Wr, const float* __restrict__ Wi,
            float* __restrict__ WrT, float* __restrict__ WiT)
{
  const int idx = blockIdx.x * blockDim.x + threadIdx.x;  // 0..2097151
  const int o = idx & 63;
  const int i = (idx >> 6) & 63;
  const int w = (idx >> 12) & 63;
  const int h = idx >> 18;
  const int src = (((h * 64 + i) * 64 + o) * 64) + w;
  const int dst = (h * 64 + w) * (E_ * 64)                // slab base (floats)
                + frag_pos(i, o, E_) * 2 + (i & 1);
  WrT[dst] = Wr[src];
  WiT[dst] = Wi[src];
}

// ---------------------------------------------------------------------------
// Kernel 1: forward DFT. One block per (b,h); 4 waves = 4 mode strips of 16.
// sel_r = Fc(64x1024) @ Q(1024x64), sel_i = Fs @ Q.
// Stored into per-(h,w) slabs in fragment order for stage 2's A operand
// (K = i, col = b).
// ---------------------------------------------------------------------------
__global__ void __launch_bounds__(128)
fwd_dft(const float* __restrict__ q, const float* __restrict__ Fc,
        const float* __restrict__ Fs, float* __restrict__ selR,
        float* __restrict__ selI)
{
  const int bh = blockIdx.x;
  const int b = bh >> 3;
  const int h = bh & 7;
  const int mt = threadIdx.x >> 5;       // mode tile 0..3
  const int lane = threadIdx.x & 31;
  const int l15 = lane & 15;
  const int kh = lane >> 4;
  const int wcol = mt * 16 + l15;        // A-matrix M index = mode

  const v2f* __restrict__ FcF = (const v2f*)Fc;
  const v2f* __restrict__ FsF = (const v2f*)Fs;

  const v8f zero = {0.f, 0.f, 0.f, 0.f, 0.f, 0.f, 0.f, 0.f};
  v8f accR[4] = {zero, zero, zero, zero};
  v8f accI[4] = {zero, zero, zero, zero};

  const float* qbase = q + ((b * L_) * H_ + h) * E_;

  for (int kb = 0; kb < L_; kb += 4) {
    const int ks = kb >> 2;
    const int l0 = kb + 2 * kh;          // K rows for j=0,1 (B operand)
    const v2f ac  = FcF[(ks * 2 + kh) * M_ + wcol];   // one b64 load
    const v2f as2 = FsF[(ks * 2 + kh) * M_ + wcol];   // one b64 load
#pragma unroll
    for (int nt = 0; nt < 4; ++nt) {
      const float* qp = qbase + l0 * (H_ * E_) + nt * 16 + l15;
      v2f bb;
      bb[0] = qp[0];
      bb[1] = qp[H_ * E_];
      accR[nt] = wmma_f32(ac,  bb, accR[nt]);
      accI[nt] = wmma_f32(as2, bb, accI[nt]);
    }
  }

#pragma unroll
  for (int nt = 0; nt < 4; ++nt) {
#pragma unroll
    for (int v = 0; v < 8; ++v) {
      const int w_ = mt * 16 + v + 8 * kh;      // C layout: M = v + 8*(lane/16)
      const int i_ = nt * 16 + l15;
      const int o = (h * M_ + w_) * (E_ * B_)   // per-(h,w) slab
                  + frag_pos(i_, b, B_) * 2 + (i_ & 1);
      selR[o] = accR[nt][v];
      selI[o] = accI[nt][v];
    }
  }
}

// ---------------------------------------------------------------------------
// Kernel 2: complex mode mixing. One block per (h,w); 4 waves = 4 b-strips.
// out_r = sel_r@Wr - sel_i@Wi ; out_i = sel_r@Wi + sel_i@Wr   (sum over i)
// fp32 WMMA cannot negate A/B (NEG=CNeg only), so keep three accumulator
// chains and subtract in VALU at the end. All operand loads are b64 fragment
// loads. Output stored into per-(b,h) slabs in fragment order for stage 3's
// B operand (K = w, col = o).
// ---------------------------------------------------------------------------
__global__ void __launch_bounds__(128)
mode_mix(const float* __restrict__ selR, const float* __restrict__ selI,
         const float* __restrict__ WrT, const float* __restrict__ WiT,
         float* __restrict__ outR, float* __restrict__ outI)
{
  const int hw = blockIdx.x;             // h*64 + w
  const int h = hw >> 6;
  const int w = hw & 63;
  const int mt = threadIdx.x >> 5;       // b tile 0..3
  const int lane = threadIdx.x & 31;
  const int l15 = lane & 15;
  const int kh = lane >> 4;

  const v2f* __restrict__ aR = (const v2f*)selR + hw * (E_ * B_ / 2);
  const v2f* __restrict__ aI = (const v2f*)selI + hw * (E_ * B_ / 2);
  const v2f* __restrict__ bR = (const v2f*)WrT  + hw * (E_ * 64 / 2);
  const v2f* __restrict__ bI = (const v2f*)WiT  + hw * (E_ * 64 / 2);

  const v8f zero = {0.f, 0.f, 0.f, 0.f, 0.f, 0.f, 0.f, 0.f};
  v8f aRR[4] = {zero, zero, zero, zero};
  v8f aII[4] = {zero, zero, zero, zero};
  v8f aMX[4] = {zero, zero, zero, zero};

  const int bcol = mt * 16 + l15;        // A-matrix M index = batch b

  for (int kb = 0; kb < E_; kb += 4) {
    const int row = (kb >> 2) * 2 + kh;
    const v2f ar = aR[row * B_ + bcol];
    const v2f ai = aI[row * B_ + bcol];
#pragma unroll
    for (int nt = 0; nt < 4; ++nt) {
      const v2f br = bR[row * E_ + nt * 16 + l15];
      const v2f bi = bI[row * E_ + nt * 16 + l15];
      aRR[nt] = wmma_f32(ar, br, aRR[nt]);
      aII[nt] = wmma_f32(ai, bi, aII[nt]);
      aMX[nt] = wmma_f32(ar, bi, aMX[nt]);
      aMX[nt] = wmma_f32(ai, br, aMX[nt]);
    }
  }

#pragma unroll
  for (int nt = 0; nt < 4; ++nt) {
#pragma unroll
    for (int v = 0; v < 8; ++v) {
      const int b_ = mt * 16 + v + 8 * kh;
      const int o_ = nt * 16 + l15;
      const int o = (b_ * H_ + h) * (M_ * E_)        // per-(b,h) slab
                  + frag_pos(w, o_, E_) * 2 + (w & 1);
      outR[o] = aRR[nt][v] - aII[nt][v];
      outI[o] = aMX[nt][v];
    }
  }
}

// ---------------------------------------------------------------------------
// Kernel 3: inverse DFT. One block per (b,h); 8 waves x 8 chunks = 64 L-tiles.
// x(1024x64) = Bc(1024x64) @ out_r + Bs(1024x64) @ out_i  (2/L factors folded
// into Bc/Bs; DC imag naturally killed since sin(0)=0). All operand loads are
// b64 fragment loads. Output written directly in (B,L,H,E) order — coalesced.
// ---------------------------------------------------------------------------
__global__ void __launch_bounds__(256)
inv_dft(const float* __restrict__ outR, const float* __restrict__ outI,
        const float* __restrict__ Bc, const float* __restrict__ Bs,
        float* __restrict__ x)
{
  const int bh = blockIdx.x;
  const int b = bh >> 3;
  const int h = bh & 7;
  const int wv = threadIdx.x >> 5;       // 0..7
  const int lane = threadIdx.x & 31;
  const int l15 = lane & 15;
  const int kh = lane >> 4;

  const v2f* __restrict__ BcF = (const v2f*)Bc;
  const v2f* __restrict__ BsF = (const v2f*)Bs;
  const v2f* __restrict__ fR = (const v2f*)outR + (b * H_ + h) * (M_ * E_ / 2);
  const v2f* __restrict__ fI = (const v2f*)outI + (b * H_ + h) * (M_ * E_ / 2);

  const v8f zero = {0.f, 0.f, 0.f, 0.f, 0.f, 0.f, 0.f, 0.f};

  for (int r = 0; r < 8; ++r) {
    const int mt = r * 8 + wv;           // L tile, 0..63
    const int lcol = mt * 16 + l15;      // A-matrix M index = l
    v8f acc[4] = {zero, zero, zero, zero};

    for (int kb = 0; kb < M_; kb += 4) {
      const int row = (kb >> 2) * 2 + kh;
      const v2f ac  = BcF[row * L_ + lcol];
      const v2f as2 = BsF[row * L_ + lcol];
#pragma unroll
      for (int nt = 0; nt < 4; ++nt) {
        const v2f br = fR[row * E_ + nt * 16 + l15];
        const v2f bi = fI[row * E_ + nt * 16 + l15];
        acc[nt] = wmma_f32(ac,  br, acc[nt]);
        acc[nt] = wmma_f32(as2, bi, acc[nt]);
      }
    }

#pragma unroll
    for (int nt = 0; nt < 4; ++nt) {
#pragma unroll
      for (int v = 0; v < 8; ++v) {
        const int l_ = mt * 16 + v + 8 * kh;
        const int e_ = nt * 16 + l15;
        x[((b * L_ + l_) * H_ + h) * E_ + e_] = acc[nt][v];
      }
    }
  }
}

// ---------------------------------------------------------------------------
extern "C" void kernel_launch(void* const* d_in, const int* in_sizes, int n_in,
                              void* d_out, int out_size, void* d_ws, size_t ws_size,
                              hipStream_t stream)
{
  const float* q  = (const float*)d_in[0];   // queries (B,L,H,E)
  // d_in[1]=keys, d_in[2]=values, d_in[3]=mask — unused by the reference.
  const float* Wr = (const float*)d_in[4];   // (H,E,E,MODES)
  const float* Wi = (const float*)d_in[5];
  float* out = (float*)d_out;                // (B,L,H,E)

  float* ws   = (float*)d_ws;
  float* Fc   = ws;                   // 65536 floats each (basis)
  float* Fs   = Fc  + 65536;
  float* Bc   = Fs  + 65536;
  float* Bs   = Bc  + 65536;
  float* WrT  = Bs  + 65536;          // 2097152 floats each below
  float* WiT  = WrT + 2097152;
  float* selR = WiT + 2097152;
  float* selI = selR + 2097152;
  float* outR = selI + 2097152;
  float* outI = outR + 2097152;       // total ~12.85M floats (~52 MB)

  build_basis<<<65536 / 256, 256, 0, stream>>>(Fc, Fs, Bc, Bs);
  transpose_w<<<2097152 / 256, 256, 0, stream>>>(Wr, Wi, WrT, WiT);
  fwd_dft<<<B_ * H_, 128, 0, stream>>>(q, Fc, Fs, selR, selI);
  mode_mix<<<H_ * M_, 128, 0, stream>>>(selR, selI, WrT, WiT, outR, outI);
  inv_dft<<<B_ * H_, 256, 0, stream>>>(outR, outI, Bc, Bs, out);
}